// ElectrochemicalInteractions_50122268344532
// MI455X (gfx1250) — compile-verified
//
#include <hip/hip_runtime.h>
#include <hip/hip_bf16.h>
#include <math.h>

// Problem constants (fixed by the reference file).
constexpr int Bb = 4;
constexpr int Nn = 2048;
constexpr int Hd = 128;
constexpr int BN = Bb * Nn;          // 8192 rows
constexpr float THRESH_FACTOR = 1.25f;

typedef __attribute__((ext_vector_type(2))) float v2f;
typedef __attribute__((ext_vector_type(8))) float v8f;

// ---------------------------------------------------------------------------
// Kernel 0: zero the f64 accumulators in workspace.
// ws layout: double S[8]  (per-batch {sum, sumsq} of dist_sq)
//            float  th[4] (per-batch threshold)
// ---------------------------------------------------------------------------
__global__ void zero_ws_kernel(double* S) {
    if (threadIdx.x < 8) S[threadIdx.x] = 0.0;
}

// ---------------------------------------------------------------------------
// Kernel 1: fused MLP + three heads using V_WMMA_F32_16X16X4_F32.
//   X = silu(coords @ W1 + b1)           (16x128 tile per wave, staged in LDS)
//   h = X @ W2 + b2                      (WMMA, 8 col-tiles x 32 K-steps)
//   d/r/m = h @ W{d,r,m} + b{d,r,m}      (epilogue: shuffle reduction, h never
//                                         touches memory)
// 128 threads = 4 waves per block; each wave owns a 16-row tile -> 64 rows
// per block, 128 blocks total.
// ---------------------------------------------------------------------------
__global__ __launch_bounds__(128) void mlp_heads_kernel(
    const float* __restrict__ coords,                  // [BN,2]
    const float* __restrict__ W1, const float* __restrict__ b1,
    const float* __restrict__ W2, const float* __restrict__ b2,
    const float* __restrict__ Wd, const float* __restrict__ bd,
    const float* __restrict__ Wr, const float* __restrict__ br,
    const float* __restrict__ Wm, const float* __restrict__ bm,
    float* __restrict__ out_d, float* __restrict__ out_r,
    float* __restrict__ out_m)
{
    __shared__ float Xs[4][16 * Hd];                   // 32 KB

    const int wave = threadIdx.x >> 5;
    const int lane = threadIdx.x & 31;
    const int half = lane >> 4;                        // 0 | 1 (lane group)
    const int l16  = lane & 15;
    const int row0 = (blockIdx.x * 4 + wave) * 16;     // first row of tile
    float* X = Xs[wave];

    // ---- Phase 1: first layer + SiLU into LDS (row r uniform per j) ----
    #pragma unroll 4
    for (int j = 0; j < 64; ++j) {
        const int r = j >> 2;
        const int k = ((j & 3) << 5) + lane;           // 0..127
        const float c0 = coords[(row0 + r) * 2 + 0];
        const float c1 = coords[(row0 + r) * 2 + 1];
        const float pre = c0 * W1[k] + c1 * W1[Hd + k] + b1[k];
        X[r * Hd + k] = pre / (1.0f + __expf(-pre));   // silu
    }
    __syncthreads();

    // ---- Phase 2: WMMA GEMM (X @ W2) + fused head epilogue ----
    float accD[8], accR[8], accM[8];
    #pragma unroll
    for (int v = 0; v < 8; ++v) { accD[v] = 0.f; accR[v] = 0.f; accM[v] = 0.f; }

    for (int n = 0; n < 8; ++n) {                      // 16-col output tiles
        const int col = n * 16 + l16;
        v8f c = {};                                    // zero accumulator
        #pragma unroll
        for (int kk = 0; kk < 32; ++kk) {              // K = 128 in steps of 4
            const int ka = kk * 4 + half * 2;
            v2f a, b;
            // A frag: M = l16, VGPR0 -> K=ka, VGPR1 -> K=ka+1 (ISA 16x4 layout)
            a.x = X[l16 * Hd + ka];
            a.y = X[l16 * Hd + ka + 1];
            // B frag: N = l16, same K striping
            b.x = W2[ka * Hd + col];
            b.y = W2[(ka + 1) * Hd + col];
            c = __builtin_amdgcn_wmma_f32_16x16x4_f32(
                    false, a, false, b, (short)0, c, false, false);
        }
        // c[v] = (X@W2)[row0 + v + 8*half, col]
        const float bb = b2[col];
        const float wd = Wd[col], wr = Wr[col], wm = Wm[col];
        #pragma unroll
        for (int v = 0; v < 8; ++v) {
            const float h = c[v] + bb;
            accD[v] += h * wd;
            accR[v] += h * wr;
            accM[v] += h * wm;
        }
    }

    // Reduce the 16 columns held by each lane-half (xor stays inside a half).
    #pragma unroll
    for (int v = 0; v < 8; ++v) {
        #pragma unroll
        for (int off = 8; off >= 1; off >>= 1) {
            accD[v] += __shfl_xor(accD[v], off, 32);
            accR[v] += __shfl_xor(accR[v], off, 32);
            accM[v] += __shfl_xor(accM[v], off, 32);
        }
    }
    const float bdv = bd[0], brv = br[0], bmv = bm[0];
    #pragma unroll
    for (int v = 0; v < 8; ++v) {
        if (l16 == v) {                                // one lane per half
            const int row = row0 + v + 8 * half;
            out_d[row] = accD[v] + bdv;
            out_r[row] = accR[v] + brv;
            out_m[row] = accM[v] + bmv;
        }
    }
}

// ---------------------------------------------------------------------------
// Kernel 2: per-batch sum / sum-of-squares of dist_sq (f64 accumulation).
// grid = (N, B); one block per source row s; coords stay L2-resident (64 KB).
// ---------------------------------------------------------------------------
__global__ __launch_bounds__(256) void dist_stats_kernel(
    const float* __restrict__ coords, double* __restrict__ S)
{
    const int b = blockIdx.y, s = blockIdx.x, tid = threadIdx.x;
    const float* cb = coords + (size_t)b * Nn * 2;
    const float rs0 = cb[s * 2 + 0];
    const float rs1 = cb[s * 2 + 1];

    double lsum = 0.0, lsq = 0.0;
    for (int t = tid; t < Nn; t += 256) {
        const float dx = rs0 * (1.0f - cb[t * 2 + 0]);
        const float dy = rs1 * (1.0f - cb[t * 2 + 1]);
        const float d2 = dx * dx + dy * dy;
        lsum += (double)d2;
        lsq  += (double)d2 * (double)d2;
    }
    __shared__ double redA[256], redB[256];
    redA[tid] = lsum; redB[tid] = lsq;
    __syncthreads();
    for (int off = 128; off > 0; off >>= 1) {
        if (tid < off) { redA[tid] += redA[tid + off]; redB[tid] += redB[tid + off]; }
        __syncthreads();
    }
    if (tid == 0) {
        atomicAdd(&S[b * 2 + 0], redA[0]);
        atomicAdd(&S[b * 2 + 1], redB[0]);
    }
}

// ---------------------------------------------------------------------------
// Kernel 3: finalize per-batch threshold = mean + 1.25 * clip(std_ddof1, 1e-6)
// ---------------------------------------------------------------------------
__global__ void finalize_thresh_kernel(const double* __restrict__ S,
                                       float* __restrict__ th)
{
    const int b = threadIdx.x;
    if (b < Bb) {
        const double M = (double)Nn * (double)Nn;
        const double mean = S[b * 2] / M;
        double var = (S[b * 2 + 1] - S[b * 2] * S[b * 2] / M) / (M - 1.0);
        if (var < 0.0) var = 0.0;
        double sd = sqrt(var);
        if (sd < 1e-6) sd = 1e-6;
        th[b] = (float)(mean + (double)THRESH_FACTOR * sd);
    }
}

// ---------------------------------------------------------------------------
// Kernel 4: attention row + normalization + diffusion_field, single pass.
// Uses cos(atan2(dy,dx)) == dx*rsqrt(d2); stages the unnormalized row in LDS
// so the 67 MB attn output is written exactly once (store-BW bound ~3us at
// 23.3 TB/s). grid = (N, B), 256 threads.
// ---------------------------------------------------------------------------
__global__ __launch_bounds__(256) void attn_kernel(
    const float* __restrict__ coords, const float* __restrict__ th,
    const float* __restrict__ diffusion,
    float* __restrict__ attn_out, float* __restrict__ dfield_out)
{
    __shared__ float row[Nn];                          // 8 KB
    __shared__ float redS[256], redD[256];

    const int b = blockIdx.y, s = blockIdx.x, tid = threadIdx.x;
    const float* cb = coords + (size_t)b * Nn * 2;
    const float* db = diffusion + (size_t)b * Nn;
    const float rs0 = cb[s * 2 + 0];
    const float rs1 = cb[s * 2 + 1];
    const float thr = th[b];

    float lsum = 0.f, ldot = 0.f;
    for (int t = tid; t < Nn; t += 256) {
        const float dx = rs0 * (1.0f - cb[t * 2 + 0]);
        const float dy = rs1 * (1.0f - cb[t * 2 + 1]);
        const float d2 = dx * dx + dy * dy;
        // 0.5*(1+cos(atan2(dy,dx)))*exp(-d2) == 0.5*(1+dx*rsqrt(d2))*exp(-d2)
        float bias;
        if (d2 > 0.0f) bias = 0.5f * (1.0f + dx * rsqrtf(d2)) * __expf(-d2);
        else           bias = 1.0f;                    // atan2(0,0)=0 path
        const float a = (d2 <= thr) ? bias : 0.0f;
        row[t] = a;
        lsum += a;
        ldot += a * db[t];
    }
    redS[tid] = lsum; redD[tid] = ldot;
    __syncthreads();
    for (int off = 128; off > 0; off >>= 1) {
        if (tid < off) { redS[tid] += redS[tid + off]; redD[tid] += redD[tid + off]; }
        __syncthreads();
    }
    const float inv = 1.0f / (redS[0] + 1e-8f);
    float* arow = attn_out + ((size_t)b * Nn + s) * Nn;
    for (int t = tid; t < Nn; t += 256) arow[t] = row[t] * inv;
    if (tid == 0) dfield_out[(size_t)b * Nn + s] = redD[0] * inv;
}

// ---------------------------------------------------------------------------
// Host side
// ---------------------------------------------------------------------------
extern "C" void kernel_launch(void* const* d_in, const int* in_sizes, int n_in,
                              void* d_out, int out_size, void* d_ws, size_t ws_size,
                              hipStream_t stream) {
    const float* coords = (const float*)d_in[0];   // [B,N,2]
    const float* W1 = (const float*)d_in[1];
    const float* b1 = (const float*)d_in[2];
    const float* W2 = (const float*)d_in[3];
    const float* b2 = (const float*)d_in[4];
    const float* Wd = (const float*)d_in[5];
    const float* bd = (const float*)d_in[6];
    const float* Wr = (const float*)d_in[7];
    const float* br = (const float*)d_in[8];
    const float* Wm = (const float*)d_in[9];
    const float* bm = (const float*)d_in[10];

    // Output layout (flat, return order):
    float* out = (float*)d_out;
    float* out_d = out;                                  // [B,N,1]
    float* out_r = out + BN;                             // [B,N,1]
    float* out_m = out + 2 * BN;                         // [B,N,1]
    float* out_a = out + 3 * BN;                         // [B,N,N]
    float* out_f = out + 3 * BN + (size_t)Bb * Nn * Nn;  // [B,N,1]

    double* S = (double*)d_ws;                           // 8 doubles
    float* th = (float*)(S + 8);                         // 4 floats

    zero_ws_kernel<<<1, 32, 0, stream>>>(S);

    mlp_heads_kernel<<<BN / 64, 128, 0, stream>>>(
        coords, W1, b1, W2, b2, Wd, bd, Wr, br, Wm, bm, out_d, out_r, out_m);

    dist_stats_kernel<<<dim3(Nn, Bb), 256, 0, stream>>>(coords, S);
    finalize_thresh_kernel<<<1, 32, 0, stream>>>(S, th);

    attn_kernel<<<dim3(Nn, Bb), 256, 0, stream>>>(coords, th, out_d, out_a, out_f);
}